// MixtureOfExpertsTreeEnsemble_56942676410674
// MI455X (gfx1250) — compile-verified
//
#include <hip/hip_runtime.h>

typedef __attribute__((ext_vector_type(2))) float v2f;
typedef __attribute__((ext_vector_type(8))) float v8f;

#define NUM_TREES   64
#define LEAF_DIMS   128
#define D_IN        512
#define N_INTERNAL  63
#define N_LEAVES    64

static __device__ __forceinline__ v8f wmma_f32(v2f a, v2f b, v8f c) {
    // D = A(16x4) * B(4x16) + C(16x16), all f32.
    // (neg_a, A, neg_b, B, c_mod, C, reuse_a, reuse_b)
    return __builtin_amdgcn_wmma_f32_16x16x4_f32(false, a, false, b,
                                                 (short)0, c, false, false);
}

static __device__ __forceinline__ float fast_sigmoid(float z) {
    return __builtin_amdgcn_rcpf(1.0f + __expf(-z));
}

// ---------------------------------------------------------------------------
// Kernel 1: wT[c][l][tt][d] = leaf_weight[l][d][t] * softmax_t(gates[l][d][:])
//           with t = c*16 + tt  (chunked-tree-major, d contiguous)
// One wave per (l,d) row; each lane handles t=lane and t=lane+32.
// ---------------------------------------------------------------------------
__global__ __launch_bounds__(256) void prep_w_kernel(
    const float* __restrict__ leaf, const float* __restrict__ gates,
    float* __restrict__ wT)
{
    const int lane = threadIdx.x & 31;
    const int wid  = threadIdx.x >> 5;
    const int row  = blockIdx.x * 8 + wid;          // (l,d) row in [0, 8192)
    const int t0 = lane, t1 = lane + 32;

    float g0 = gates[(size_t)row * 64 + t0];
    float g1 = gates[(size_t)row * 64 + t1];

    float m = fmaxf(g0, g1);
#pragma unroll
    for (int off = 16; off >= 1; off >>= 1)
        m = fmaxf(m, __shfl_xor(m, off, 32));

    float e0 = __expf(g0 - m), e1 = __expf(g1 - m);
    float s = e0 + e1;
#pragma unroll
    for (int off = 16; off >= 1; off >>= 1)
        s += __shfl_xor(s, off, 32);
    const float inv = 1.0f / s;

    const int l = row >> 7;          // leaf index
    const int d = row & 127;         // leaf dim
    float w0 = leaf[(size_t)row * 64 + t0] * e0 * inv;
    float w1 = leaf[(size_t)row * 64 + t1] * e1 * inv;

    // layout: c*131072 + l*2048 + tt*128 + d
    wT[(size_t)(t0 >> 4) * 131072 + (size_t)l * 2048 + (size_t)(t0 & 15) * 128 + d] = w0;
    wT[(size_t)(t1 >> 4) * 131072 + (size_t)l * 2048 + (size_t)(t1 & 15) * 128 + d] = w1;
}

// ---------------------------------------------------------------------------
// Kernel 2: fused tree-ensemble. One workgroup = 16 batch rows, 8 waves.
// ---------------------------------------------------------------------------
__global__ __launch_bounds__(256) void tree_main_kernel(
    const float* __restrict__ x, const float* __restrict__ W,
    const float* __restrict__ bias, const float* __restrict__ wT,
    float* __restrict__ out)
{
    extern __shared__ float smem[];
    float* xT  = smem;                    //  8192 f32: xT[d*16 + m]
    float* gld = smem + 8192;             // 16128 f32: gld[n*256 + row*16 + tt]
    float* pld = smem + 8192 + 16128;     // 16384 f32: pld[k*16 + m], k = l*16 + tt

    const int tid  = threadIdx.x;
    const int lane = tid & 31;
    const int wid  = tid >> 5;            // wave 0..7
    const int half = lane >> 4;           // 0/1  (K-half for A/B fragments)
    const int q    = lane & 15;           // row/col-within-16
    const int blk  = blockIdx.x;          // batch tile of 16 rows

    // ---- stage x tile, transposed for conflict-free A-fragment reads ----
    for (int idx = tid; idx < 16 * 512; idx += 256) {
        int m = idx >> 9;
        int d = idx & 511;
        xT[d * 16 + m] = x[(size_t)(blk * 16 + m) * 512 + d];
    }
    __syncthreads();

    // two independent output accumulator chains (summed at the end)
    v8f accA = {0.f, 0.f, 0.f, 0.f, 0.f, 0.f, 0.f, 0.f};
    v8f accB = {0.f, 0.f, 0.f, 0.f, 0.f, 0.f, 0.f, 0.f};

    const int nbeg = wid * 8;
    const int nend = (nbeg + 8 < N_INTERNAL) ? nbeg + 8 : N_INTERNAL;

    for (int c = 0; c < 4; ++c) {
        const int t0 = c * 16;

        // ---- Phase A: logits + sigmoid; node PAIRS share the A fragment ----
        int n = nbeg;
        for (; n + 1 < nend; n += 2) {
            const float bb0 = bias[n * 64 + t0 + q];          // in flight
            const float bb1 = bias[(n + 1) * 64 + t0 + q];    // during GEMM
            v8f acc0 = {0.f, 0.f, 0.f, 0.f, 0.f, 0.f, 0.f, 0.f};
            v8f acc1 = {0.f, 0.f, 0.f, 0.f, 0.f, 0.f, 0.f, 0.f};
            const float* wp0 = W + (size_t)n * 32768 + (size_t)(t0 + q); // + d*64
            const float* wp1 = wp0 + 32768;
#pragma unroll 4
            for (int kb = 0; kb < 512; kb += 4) {
                const int d0 = kb + half * 2;
                v2f a, b0, b1;
                a.x  = xT[d0 * 16 + q];
                a.y  = xT[(d0 + 1) * 16 + q];
                b0.x = wp0[(size_t)d0 * 64];
                b0.y = wp0[(size_t)(d0 + 1) * 64];
                b1.x = wp1[(size_t)d0 * 64];
                b1.y = wp1[(size_t)(d0 + 1) * 64];
                acc0 = wmma_f32(a, b0, acc0);
                acc1 = wmma_f32(a, b1, acc1);
            }
#pragma unroll
            for (int r = 0; r < 8; ++r) {
                const int rr = (r + half * 8) * 16 + q;
                gld[n * 256 + rr]       = fast_sigmoid(acc0[r] + bb0);
                gld[(n + 1) * 256 + rr] = fast_sigmoid(acc1[r] + bb1);
            }
        }
        if (n < nend) {   // odd tail (wave 7, node 62)
            const float bb = bias[n * 64 + t0 + q];
            v8f acc = {0.f, 0.f, 0.f, 0.f, 0.f, 0.f, 0.f, 0.f};
            const float* wp = W + (size_t)n * 32768 + (size_t)(t0 + q);
#pragma unroll 4
            for (int kb = 0; kb < 512; kb += 4) {
                const int d0 = kb + half * 2;
                v2f a, bm;
                a.x  = xT[d0 * 16 + q];
                a.y  = xT[(d0 + 1) * 16 + q];
                bm.x = wp[(size_t)d0 * 64];
                bm.y = wp[(size_t)(d0 + 1) * 64];
                acc = wmma_f32(a, bm, acc);
            }
#pragma unroll
            for (int r = 0; r < 8; ++r)
                gld[n * 256 + (r + half * 8) * 16 + q] = fast_sigmoid(acc[r] + bb);
        }
        __syncthreads();

        // ---- Phase B: tree probability propagation (in-place, descending) ----
        {
            const int prow = tid & 15;     // batch row
            const int ttg  = tid >> 4;     // tree within chunk
#define PSLOT(l) pld[(l) * 256 + ttg * 16 + prow]
            PSLOT(0) = 1.0f;
#pragma unroll
            for (int lvl = 0; lvl < 6; ++lvl) {
                const int nlvl  = 1 << lvl;
                const int start = nlvl - 1;
                for (int i = nlvl - 1; i >= 0; --i) {
                    float par = PSLOT(i);
                    float g = gld[(start + i) * 256 + prow * 16 + ttg];
                    PSLOT(2 * i + 1) = par * (1.0f - g);
                    PSLOT(2 * i)     = par * g;
                }
            }
#undef PSLOT
        }
        __syncthreads();

        // ---- Phase C: out += p(16 x 1024) @ wT_chunk(1024 x 128) ----
        // two interleaved accumulator chains for XDL pipelining
        {
            const float* wp = wT + (size_t)c * 131072 + (size_t)(wid * 16 + q); // + k*128
#pragma unroll 2
            for (int kb = 0; kb < 1024; kb += 8) {
                const int k0 = kb + half * 2;
                const int k1 = k0 + 4;
                v2f a0, b0, a1, b1;
                a0.x = pld[k0 * 16 + q];
                a0.y = pld[(k0 + 1) * 16 + q];
                b0.x = wp[(size_t)k0 * 128];
                b0.y = wp[(size_t)(k0 + 1) * 128];
                a1.x = pld[k1 * 16 + q];
                a1.y = pld[(k1 + 1) * 16 + q];
                b1.x = wp[(size_t)k1 * 128];
                b1.y = wp[(size_t)(k1 + 1) * 128];
                accA = wmma_f32(a0, b0, accA);
                accB = wmma_f32(a1, b1, accB);
            }
        }
        // next iteration's Phase A writes gld (done being read); pld is only
        // rewritten after the A->B barrier, so no extra barrier needed here.
    }

    // ---- write 16x128 output tile ----
#pragma unroll
    for (int r = 0; r < 8; ++r) {
        const int row = r + half * 8;
        out[(size_t)(blk * 16 + row) * 128 + wid * 16 + q] = accA[r] + accB[r];
    }
}

// ---------------------------------------------------------------------------
extern "C" void kernel_launch(void* const* d_in, const int* in_sizes, int n_in,
                              void* d_out, int out_size, void* d_ws, size_t ws_size,
                              hipStream_t stream) {
    const float* x     = (const float*)d_in[0];   // 4096 x 512
    const float* W     = (const float*)d_in[1];   // 63 x 512 x 64
    const float* b     = (const float*)d_in[2];   // 63 x 64
    const float* leaf  = (const float*)d_in[3];   // 64 x 128 x 64
    const float* gates = (const float*)d_in[4];   // 64 x 128 x 64
    float* out = (float*)d_out;                   // 4096 x 128
    float* wT  = (float*)d_ws;                    // 2 MB scratch

    prep_w_kernel<<<1024, 256, 0, stream>>>(leaf, gates, wT);

    const size_t lds_bytes = (size_t)(8192 + 16128 + 16384) * sizeof(float); // ~159 KB
    tree_main_kernel<<<256, 256, lds_bytes, stream>>>(x, W, b, wT, out);
}